// ModelFull_78245714198552
// MI455X (gfx1250) — compile-verified
//
#include <hip/hip_runtime.h>

typedef __attribute__((ext_vector_type(2))) float v2f;
typedef __attribute__((ext_vector_type(8))) float v8f;

#define FEATS   128
#define CLASSES 64

// ---------------------------------------------------------------------------
// Kernel 1: sum[v][:] = x[v][:]  (numerator starts at x[v]),  deg[v] = 1.0
// (so after scatter:  h = sum/deg == (neigh_sum + x)/(in_deg + 1))
// ---------------------------------------------------------------------------
__global__ void sage_init_kernel(const float* __restrict__ x,
                                 float* __restrict__ sum,
                                 float* __restrict__ deg,
                                 int n_nodes) {
    size_t i = (size_t)blockIdx.x * blockDim.x + threadIdx.x;
    size_t total = (size_t)n_nodes * FEATS;
    if (i < total) sum[i] = x[i];
    if (i < (size_t)n_nodes) deg[i] = 1.0f;
}

// ---------------------------------------------------------------------------
// Kernel 2: edge scatter. One wave (32 lanes) per edge; each lane moves a
// float4 (128 feats / 32 lanes = 4). x rows are L2-resident (51.2MB < 192MB),
// atomics are native global_atomic_add_f32 without return (STOREcnt path).
// ---------------------------------------------------------------------------
__global__ void sage_scatter_kernel(const float* __restrict__ x,
                                    const int* __restrict__ esrc,
                                    const int* __restrict__ edst,
                                    float* __restrict__ sum,
                                    float* __restrict__ deg,
                                    int n_edges) {
    int t = blockIdx.x * blockDim.x + threadIdx.x;
    int e = t >> 5;
    int lane = t & 31;
    if (e >= n_edges) return;
    int s = esrc[e];
    int d = edst[e];
    const float4 v = *(const float4*)(x + (size_t)s * FEATS + lane * 4);
    float* dp = sum + (size_t)d * FEATS + lane * 4;
    __hip_atomic_fetch_add(dp + 0, v.x, __ATOMIC_RELAXED, __HIP_MEMORY_SCOPE_AGENT);
    __hip_atomic_fetch_add(dp + 1, v.y, __ATOMIC_RELAXED, __HIP_MEMORY_SCOPE_AGENT);
    __hip_atomic_fetch_add(dp + 2, v.z, __ATOMIC_RELAXED, __HIP_MEMORY_SCOPE_AGENT);
    __hip_atomic_fetch_add(dp + 3, v.w, __ATOMIC_RELAXED, __HIP_MEMORY_SCOPE_AGENT);
    if (lane == 0)
        __hip_atomic_fetch_add(deg + d, 1.0f, __ATOMIC_RELAXED, __HIP_MEMORY_SCOPE_AGENT);
}

// ---------------------------------------------------------------------------
// Kernel 3: out = (sum/deg) @ W^T + b   via V_WMMA_F32_16X16X4_F32.
// One wave owns a 16-node x 64-class tile: 4 accumulators (v8f) over the
// N dimension, K stepped by 4 -> 32 iterations x 4 WMMA each.
//
// W staged into LDS in PAIR-INTERLEAVED B-fragment order:
//   w_lds[(k>>1)*128 + n*2 + (k&1)] = W[n][k]
// so each lane's B fragment {W[n][K], W[n][K+1]} is one aligned 8-byte LDS
// load (no cross-VGPR repacking movs).  K = k + khalf is always even
// (k steps by 4, khalf in {0,2}), so the (even,odd) pair is exact.
//
// Fragment layouts (ISA 7.12.2, 32-bit, wave32):
//   A 16x4:  lane L(<16): M=L holds K=0(v0),K=1(v1); lane 16+L: K=2,K=3
//   B 4x16:  v0: K=0 (lanes 0-15, N=lane) / K=2 (lanes 16-31); v1: K=1/K=3
//   D 16x16: VGPR r: lanes 0-15 -> M=r, N=lane; lanes 16-31 -> M=r+8
// ---------------------------------------------------------------------------
__global__ void __launch_bounds__(256)
sage_gemm_wmma_kernel(const float* __restrict__ sum,
                      const float* __restrict__ deg,
                      const float* __restrict__ W,     // [CLASSES][FEATS]
                      const float* __restrict__ bias,  // [CLASSES]
                      float* __restrict__ out,         // [n][CLASSES]
                      int n_tiles) {
    __shared__ float w_lds[FEATS * CLASSES]; // pair-interleaved, see header

    int tid = threadIdx.x;
    for (int i = tid; i < FEATS * CLASSES; i += 256) {
        int k = i & (FEATS - 1);   // fast index -> coalesced global read
        int n = i >> 7;
        w_lds[(k >> 1) * 128 + n * 2 + (k & 1)] = W[n * FEATS + k];
    }
    __syncthreads();

    const int lane  = tid & 31;
    const int wave  = tid >> 5;
    const int mrow  = lane & 15;          // M (and N) sub-index for this lane
    const int khalf = (lane >> 4) << 1;   // 0 for lanes 0-15, 2 for 16-31

    const int gwave  = blockIdx.x * (blockDim.x >> 5) + wave;
    const int nwaves = gridDim.x * (blockDim.x >> 5);

    const float bias0 = bias[ 0 + mrow];
    const float bias1 = bias[16 + mrow];
    const float bias2 = bias[32 + mrow];
    const float bias3 = bias[48 + mrow];

    // per-lane base into the pair-interleaved LDS image (dword units)
    const float* wl = &w_lds[(khalf >> 1) * 128 + mrow * 2];

    for (int tile = gwave; tile < n_tiles; tile += nwaves) {
        const int row  = tile * 16 + mrow;
        const float invd = 1.0f / deg[row];
        const float* hrow = sum + (size_t)row * FEATS + khalf;

        v8f acc0 = {}, acc1 = {}, acc2 = {}, acc3 = {};

        #pragma unroll 4
        for (int k = 0; k < FEATS; k += 4) {
            float2 hv = *(const float2*)(hrow + k);         // K=k+khalf, +1
            v2f a = { hv.x * invd, hv.y * invd };           // fold 1/deg into A

            // pair-row index for K = k+khalf is (k+khalf)/2 = k/2 + khalf/2
            const float* wk = wl + (k >> 1) * 128;
            v2f b0 = *(const v2f*)(wk +  0);   // n = mrow
            v2f b1 = *(const v2f*)(wk + 32);   // n = mrow+16
            v2f b2 = *(const v2f*)(wk + 64);   // n = mrow+32
            v2f b3 = *(const v2f*)(wk + 96);   // n = mrow+48

            acc0 = __builtin_amdgcn_wmma_f32_16x16x4_f32(false, a, false, b0, (short)0, acc0, false, false);
            acc1 = __builtin_amdgcn_wmma_f32_16x16x4_f32(false, a, false, b1, (short)0, acc1, false, false);
            acc2 = __builtin_amdgcn_wmma_f32_16x16x4_f32(false, a, false, b2, (short)0, acc2, false, false);
            acc3 = __builtin_amdgcn_wmma_f32_16x16x4_f32(false, a, false, b3, (short)0, acc3, false, false);
        }

        // D layout: element r -> M = r (+8 for upper lane half), N = mrow
        const int rbase = tile * 16 + ((lane >> 4) << 3);
        #pragma unroll
        for (int r = 0; r < 8; ++r) {
            float* op = out + (size_t)(rbase + r) * CLASSES + mrow;
            op[ 0] = acc0[r] + bias0;
            op[16] = acc1[r] + bias1;
            op[32] = acc2[r] + bias2;
            op[48] = acc3[r] + bias3;
        }
    }
}

// ---------------------------------------------------------------------------
extern "C" void kernel_launch(void* const* d_in, const int* in_sizes, int n_in,
                              void* d_out, int out_size, void* d_ws, size_t ws_size,
                              hipStream_t stream) {
    const float* x    = (const float*)d_in[0];
    const int*   esrc = (const int*)d_in[1];
    const int*   edst = (const int*)d_in[2];
    const float* W    = (const float*)d_in[3];
    const float* b    = (const float*)d_in[4];
    float* out = (float*)d_out;

    const int n_nodes = in_sizes[0] / FEATS;   // 100000
    const int n_edges = in_sizes[1];           // 800000

    float* sum = (float*)d_ws;                          // n_nodes*128 floats
    float* deg = sum + (size_t)n_nodes * FEATS;         // n_nodes floats

    {   // init: sum = x, deg = 1
        size_t total = (size_t)n_nodes * FEATS;
        int blocks = (int)((total + 255) / 256);
        sage_init_kernel<<<blocks, 256, 0, stream>>>(x, sum, deg, n_nodes);
    }
    {   // edge scatter: one wave per edge
        long long threads = (long long)n_edges * 32;
        int blocks = (int)((threads + 255) / 256);
        sage_scatter_kernel<<<blocks, 256, 0, stream>>>(x, esrc, edst, sum, deg, n_edges);
    }
    {   // WMMA GEMM: 16-row tiles, 8 waves per block, one tile per wave
        int n_tiles = n_nodes / 16;            // 6250 (exact: 100000 = 16*6250)
        int blocks = (n_tiles + 7) / 8;
        sage_gemm_wmma_kernel<<<blocks, 256, 0, stream>>>(sum, deg, W, b, out, n_tiles);
    }
}